// StereoGraphNet_45921790329545
// MI455X (gfx1250) — compile-verified
//
#include <hip/hip_runtime.h>
#include <hip/hip_bf16.h>

typedef float v2f __attribute__((ext_vector_type(2)));
typedef float v8f __attribute__((ext_vector_type(8)));

#define MAXSEG 512   // compile-time LDS cap (N = 400 at runtime)
#define CB 8         // channels per workgroup (1 channel per wave, 8 waves)
#define PCHUNK 8192  // pixels per workgroup (divides P = 524288)

// ---------------------------------------------------------------------------
// Zero the workspace accumulators (sums[N*C] + counts[N]) every launch.
// ---------------------------------------------------------------------------
__global__ void zero_ws_kernel(float* ws, int n) {
    int i = blockIdx.x * blockDim.x + threadIdx.x;
    if (i < n) ws[i] = 0.0f;
}

// ---------------------------------------------------------------------------
// Per-segment pixel counts: LDS histogram per workgroup, then global flush.
// ---------------------------------------------------------------------------
__global__ void count_kernel(const int* __restrict__ seg, float* __restrict__ counts,
                             int P, int N, int chunk) {
    __shared__ int hist[MAXSEG];
    for (int i = threadIdx.x; i < N; i += blockDim.x) hist[i] = 0;
    __syncthreads();
    int pstart = blockIdx.x * chunk;
    for (int i = threadIdx.x; i < chunk; i += blockDim.x) {
        int g = pstart + i;
        if (g < P) atomicAdd(&hist[seg[g]], 1);
    }
    __syncthreads();
    for (int i = threadIdx.x; i < N; i += blockDim.x) {
        int c = hist[i];
        if (c) atomicAdd(&counts[i], (float)c);
    }
}

// ---------------------------------------------------------------------------
// Streaming scatter-reduce. Grid: (pixel blocks, channel blocks of CB).
// Each wave owns one channel plane. Hot loop is 4-wide vectorized:
// global_load_b128 (512B/wave/iter) + ds_load_b128 (seg) + 4x ds_add_f32,
// to keep per-SIMD issue bandwidth well above the 23.3 TB/s HBM requirement.
// seg slice is staged once into LDS and reused by all 8 waves.
// ---------------------------------------------------------------------------
__global__ void accum_kernel(const float* __restrict__ feat,
                             const int* __restrict__ seg,
                             float* __restrict__ sums,
                             int P, int C, int N, int chunk) {
    __shared__ int   lseg[PCHUNK];
    __shared__ float lsums[CB * MAXSEG];

    // zero local accumulators
    for (int i = threadIdx.x; i < CB * N; i += blockDim.x) lsums[i] = 0.0f;

    // stage segment ids for this pixel chunk (vectorized b128 when full chunk)
    int pstart = blockIdx.x * chunk;
    int navail = P - pstart;
    int n = (navail < chunk) ? navail : chunk;
    if (n == chunk) {
        const int4* seg4 = (const int4*)(seg + pstart);
        int4* lseg4 = (int4*)lseg;
        for (int i = threadIdx.x; i < chunk / 4; i += blockDim.x)
            lseg4[i] = seg4[i];
    } else {
        for (int i = threadIdx.x; i < chunk; i += blockDim.x)
            lseg[i] = (i < n) ? seg[pstart + i] : 0;
    }
    __syncthreads();

    int wave = threadIdx.x >> 5;
    int lane = threadIdx.x & 31;
    int c = blockIdx.y * CB + wave;
    const float* plane = feat + (size_t)c * (size_t)P + pstart;
    float* lrow = &lsums[wave * N];

    if (n == chunk) {
        // fast path: 4 pixels per lane per iteration, 128 pixels per wave-iter
        const float4* plane4 = (const float4*)plane;
        const int4*   lseg4  = (const int4*)lseg;
        for (int i = lane; i < chunk / 4; i += 32) {
            if (i + 256 < chunk / 4)  // gfx1250 global_prefetch_b8 look-ahead
                __builtin_prefetch(&plane4[i + 256], 0, 0);
            float4 v = plane4[i];
            int4   s = lseg4[i];
            atomicAdd(&lrow[s.x], v.x);   // ds_add_f32
            atomicAdd(&lrow[s.y], v.y);
            atomicAdd(&lrow[s.z], v.z);
            atomicAdd(&lrow[s.w], v.w);
        }
    } else {
        for (int i = lane; i < n; i += 32) {
            float v = plane[i];
            atomicAdd(&lrow[lseg[i]], v);
        }
    }
    __syncthreads();

    // flush private accumulators to global (skip exact zeros)
    for (int i = threadIdx.x; i < CB * N; i += blockDim.x) {
        float v = lsums[i];
        if (v != 0.0f) {
            int w = i / N;
            int s = i - w * N;
            atomicAdd(&sums[(size_t)s * C + blockIdx.y * CB + w], v);
        }
    }
}

// ---------------------------------------------------------------------------
// Finalize: out = diag(inv_count) x sums via V_WMMA_F32_16X16X4_F32.
// inv_count[n] = count>0 ? 1/count : 0  -> empty segments become zero rows,
// matching the reference's where(counts>0, sums/max(counts,1), 0).
// Block = (C/16) waves; wave w handles channel tile c0 = 16*w; blockIdx.x is
// the 16-row segment tile. N=400, C=128 divide exactly.
// ---------------------------------------------------------------------------
__global__ void finalize_wmma_kernel(const float* __restrict__ sums,
                                     const float* __restrict__ counts,
                                     float* __restrict__ out,
                                     int C, int N) {
    int wave = threadIdx.x >> 5;
    int lane = threadIdx.x & 31;
    int half = lane >> 4;      // 0: lanes 0-15, 1: lanes 16-31
    int m    = lane & 15;      // row within tile (A/C/D), col within tile (B/D)
    int n0   = blockIdx.x * 16;
    int c0   = wave * 16;

    float cnt = counts[n0 + m];
    float inv = (cnt > 0.0f) ? (1.0f / cnt) : 0.0f;

    v8f acc = {};
#pragma unroll
    for (int k0 = 0; k0 < 16; k0 += 4) {
        // A (16x4 f32): VGPR0 holds K = 2*half, VGPR1 holds K = 2*half+1
        int kx = k0 + 2 * half;
        v2f a, b;
        a.x = (m == kx)     ? inv : 0.0f;   // diag(inv_count) chunk
        a.y = (m == kx + 1) ? inv : 0.0f;
        // B (4x16 f32): VGPR0 row K = 2*half, VGPR1 row K = 2*half+1; col = m
        b.x = sums[(size_t)(n0 + kx)     * C + c0 + m];
        b.y = sums[(size_t)(n0 + kx + 1) * C + c0 + m];
        acc = __builtin_amdgcn_wmma_f32_16x16x4_f32(
            /*neg_a=*/false, a, /*neg_b=*/false, b,
            /*c_mod=*/(short)0, acc, /*reuse_a=*/false, /*reuse_b=*/false);
    }

    // D (16x16 f32): VGPR v -> row M = v + 8*half, col N = m
#pragma unroll
    for (int v = 0; v < 8; ++v) {
        out[(size_t)(n0 + v + 8 * half) * C + c0 + m] = acc[v];
    }
}

// ---------------------------------------------------------------------------
extern "C" void kernel_launch(void* const* d_in, const int* in_sizes, int n_in,
                              void* d_out, int out_size, void* d_ws, size_t ws_size,
                              hipStream_t stream) {
    const float* feat = (const float*)d_in[0];   // [C, Hf, Wf] f32
    const int*   seg  = (const int*)d_in[1];     // [Hf*Wf] i32
    int P = in_sizes[1];                         // 524288 pixels
    int C = in_sizes[0] / P;                     // 128 channels
    int N = out_size / C;                        // 400 segments

    float* sums   = (float*)d_ws;                // [N*C]
    float* counts = sums + (size_t)N * C;        // [N]
    float* out    = (float*)d_out;

    int tot = N * C + N;
    zero_ws_kernel<<<(tot + 255) / 256, 256, 0, stream>>>(sums, tot);

    int chunk   = PCHUNK;
    int pblocks = (P + chunk - 1) / chunk;       // 64

    count_kernel<<<pblocks, 256, 0, stream>>>(seg, counts, P, N, chunk);

    dim3 agrid(pblocks, C / CB);                 // 64 x 16 workgroups
    accum_kernel<<<agrid, 256, 0, stream>>>(feat, seg, sums, P, C, N, chunk);

    finalize_wmma_kernel<<<N / 16, (C / 16) * 32, 0, stream>>>(sums, counts, out, C, N);
}